// trackerNet_14267881358025
// MI455X (gfx1250) — compile-verified
//
#include <hip/hip_runtime.h>

typedef float v2f __attribute__((ext_vector_type(2)));
typedef float v8f __attribute__((ext_vector_type(8)));

#define NT1 1024   // frame-t nodes (src)
#define NT2 1024   // frame-t+1 nodes (dst)
#define NN  2048   // total nodes
#define DIN 6
#define HD  64
#define CCH 3

// ---------------------------------------------------------------------------
// Kernel 1: h = relu(x @ W1 + b1)   [2048 x 64], K=6 -> scalar FMA is optimal
// ---------------------------------------------------------------------------
__global__ void tk_node_mlp(const float* __restrict__ x,
                            const float* __restrict__ W1,
                            const float* __restrict__ b1,
                            float* __restrict__ h_out) {
  int id = blockIdx.x * blockDim.x + threadIdx.x;   // NN*HD threads
  int n = id >> 6;
  int j = id & 63;
  float acc = b1[j];
#pragma unroll
  for (int k = 0; k < DIN; ++k)
    acc = fmaf(x[n * DIN + k], W1[k * HD + j], acc);
  h_out[id] = fmaxf(acc, 0.0f);
}

// ---------------------------------------------------------------------------
// Kernel 2: A = h @ W2  (+ b2 folded into rows < NT1)
// M=2048 N=64 K=64, one 16x16 tile per wave via V_WMMA_F32_16X16X4_F32
// A fragment: lane L holds A[M=L%16][K=2*(L/16)+v] ; B: lane L holds
// B[K=v+2*(L/16)][N=L%16] ; C: vgpr v -> M=v+8*(L/16), N=L%16.
// ---------------------------------------------------------------------------
__global__ void tk_hW2_wmma(const float* __restrict__ h,
                            const float* __restrict__ W2,
                            const float* __restrict__ b2,
                            float* __restrict__ A) {
  int wid  = (blockIdx.x * blockDim.x + threadIdx.x) >> 5;  // 512 waves total
  int lane = threadIdx.x & 31;
  int r  = lane & 15;     // M row (A) / N col (B,C)
  int hf = lane >> 4;     // lane half
  int tm = (wid >> 2) << 4;   // 128 M-tiles
  int tn = (wid & 3) << 4;    // 4 N-tiles

  v8f c = {};
#pragma unroll
  for (int ks = 0; ks < 16; ++ks) {
    int k = 4 * ks + 2 * hf;
    v2f a = *(const v2f*)(h + (tm + r) * HD + k);   // contiguous K pair
    v2f b;
    b.x = W2[k * HD + tn + r];
    b.y = W2[(k + 1) * HD + tn + r];
    c = __builtin_amdgcn_wmma_f32_16x16x4_f32(false, a, false, b,
                                              (short)0, c, false, false);
  }
  // fold b2 into the src-node rows (tiles are 16-aligned, NT1 is 16-aligned)
  float bias = (tm < NT1) ? b2[tn + r] : 0.0f;
#pragma unroll
  for (int v = 0; v < 8; ++v)
    A[(tm + v + 8 * hf) * HD + tn + r] = c[v] + bias;
}

// ---------------------------------------------------------------------------
// Kernel 3: scores[i,j,:] = relu(A[i]-A'[j]) @ W3 + b3
// Wave handles a 16-dst tile x 16 src rows. The WMMA A-fragment for edge
// group (i, j0..j0+15) is computed IN-LAYOUT: M=lane%16 -> dst j (lane-
// varying d), K depends only on lane/16 -> src s is a broadcast load.
// d-fragments + padded-W3 B-fragments are register-resident across all 16 i.
// ---------------------------------------------------------------------------
__global__ void tk_edge_wmma(const float* __restrict__ A,
                             const float* __restrict__ W3,
                             const float* __restrict__ b3,
                             float* __restrict__ scores) {
  int lane = threadIdx.x & 31;
  int wave = threadIdx.x >> 5;           // 8 waves / block
  int r  = lane & 15;
  int hf = lane >> 4;
  int jt = blockIdx.x & 63;              // 64 dst tiles
  int ib = blockIdx.x >> 6;              // 8 src blocks
  int j0 = jt << 4;
  int i0 = ib * 128 + wave * 16;

  const float* Asrc = A;                 // rows [0, NT1)   (b2 folded in)
  const float* Adst = A + NT1 * HD;      // rows [NT1, NN)

  // hoist dst fragments: 32 floats/lane, reused for 16 src rows
  float dreg[32];
#pragma unroll
  for (int ks = 0; ks < 16; ++ks) {
    v2f t = *(const v2f*)(Adst + (j0 + r) * HD + 4 * ks + 2 * hf);
    dreg[2 * ks] = t.x;
    dreg[2 * ks + 1] = t.y;
  }
  // hoist B fragments: W3 zero-padded 64x3 -> 64x16
  v2f bfr[16];
#pragma unroll
  for (int ks = 0; ks < 16; ++ks) {
    int k = 4 * ks + 2 * hf;
    v2f b; b.x = 0.0f; b.y = 0.0f;
    if (r < CCH) { b.x = W3[k * CCH + r]; b.y = W3[(k + 1) * CCH + r]; }
    bfr[ks] = b;
  }
  float bias = (r < CCH) ? b3[r] : 0.0f;

  for (int ii = 0; ii < 16; ++ii) {
    int i = i0 + ii;
    float sreg[32];                      // broadcast loads (L2-resident A)
#pragma unroll
    for (int ks = 0; ks < 16; ++ks) {
      v2f t = *(const v2f*)(Asrc + i * HD + 4 * ks + 2 * hf);
      sreg[2 * ks] = t.x;
      sreg[2 * ks + 1] = t.y;
    }
    v8f c = {};
#pragma unroll
    for (int ks = 0; ks < 16; ++ks) {
      v2f a;
      a.x = fmaxf(sreg[2 * ks]     - dreg[2 * ks],     0.0f);
      a.y = fmaxf(sreg[2 * ks + 1] - dreg[2 * ks + 1], 0.0f);
      c = __builtin_amdgcn_wmma_f32_16x16x4_f32(false, a, false, bfr[ks],
                                                (short)0, c, false, false);
    }
    if (r < CCH) {                       // only channels 0..2 are real
#pragma unroll
      for (int v = 0; v < 8; ++v) {
        int j = j0 + v + 8 * hf;         // C row M -> dst offset
        scores[(i * NT2 + j) * CCH + r] = c[v] + bias;
      }
    }
  }
}

// ---------------------------------------------------------------------------
extern "C" void kernel_launch(void* const* d_in, const int* in_sizes, int n_in,
                              void* d_out, int out_size, void* d_ws, size_t ws_size,
                              hipStream_t stream) {
  const float* x  = (const float*)d_in[0];
  const float* W1 = (const float*)d_in[1];
  const float* b1 = (const float*)d_in[2];
  const float* W2 = (const float*)d_in[3];
  const float* b2 = (const float*)d_in[4];
  const float* W3 = (const float*)d_in[5];
  const float* b3 = (const float*)d_in[6];
  // d_in[7] = edge_index (dense bipartite, pattern is static), d_in[8,9] = node counts

  float* out    = (float*)d_out;
  float* h_out  = out;                  // output 0: h  [2048*64]
  float* scores = out + NN * HD;        // output 1: scores [1024*1024*3]
  float* ws_A   = (float*)d_ws;         // A = h@W2 (+b2 on src rows), 512 KB

  tk_node_mlp<<<(NN * HD) / 256, 256, 0, stream>>>(x, W1, b1, h_out);
  tk_hW2_wmma<<<64, 256, 0, stream>>>(h_out, W2, b2, ws_A);
  tk_edge_wmma<<<512, 256, 0, stream>>>(ws_A, W3, b3, scores);
}